// NETWORK_OF_DANS_16338055594193
// MI455X (gfx1250) — compile-verified
//
#include <hip/hip_runtime.h>
#include <math.h>
#include <stdint.h>

typedef float v2f __attribute__((ext_vector_type(2)));
typedef float v4f __attribute__((ext_vector_type(4)));
typedef float v8f __attribute__((ext_vector_type(8)));

#define NCOL   65536   // D1*C == D2*C
#define COLS4  16384   // NCOL / 4
#define KSPLIT 8
#define KCHUNK 512     // 4096 / KSPLIT

__device__ __forceinline__ float leaky(float v) { return v > 0.f ? v : 0.01f * v; }

// -------- K0: h = x*W0 + b0  (65536 elementwise, vec4) --------
__global__ void k_affine1(const float* __restrict__ x, const float* __restrict__ W0,
                          const float* __restrict__ b0, float* __restrict__ h) {
  int i = blockIdx.x * blockDim.x + threadIdx.x;   // 0..16383
  float xv = x[0];
  v4f w = ((const v4f*)W0)[i];
  v4f b = ((const v4f*)b0)[i];
  ((v4f*)h)[i] = xv * w + b;
}

// -------- batched DAN via V_WMMA_F32_16X16X4_F32 --------------
// h: [4096*16] f32; out: a[4096]. One wave32 = one 16-row tile.
__global__ void k_dan_batch(const float* __restrict__ h, float* __restrict__ aout,
                            const float* __restrict__ Wd1, const float* __restrict__ bd1,
                            const float* __restrict__ Wd2, const float* __restrict__ bd2,
                            const float* __restrict__ Wd3, const float* __restrict__ bd3,
                            int code) {
  __shared__ float sW1[20][16];       // Wd1 padded [19x15 -> 20x16]
  __shared__ float sW2[16][16];       // Wd2 padded [15x8  -> 16x16]
  __shared__ float sW3[8];
  __shared__ float sb1[16];
  __shared__ float sb2[16];
  __shared__ float zs[8][16][20];     // per-wave z tile
  __shared__ float hsm[8][16][16];    // per-wave intermediate (layout bounce)

  const int tid = threadIdx.x;
  for (int i = tid; i < 320; i += 256) {
    int k = i >> 4, n = i & 15;
    sW1[k][n] = (k < 19 && n < 15) ? Wd1[k * 15 + n] : 0.f;
  }
  { int k = tid >> 4, n = tid & 15;
    sW2[k][n] = (k < 15 && n < 8) ? Wd2[k * 8 + n] : 0.f; }
  if (tid < 8)  sW3[tid] = Wd3[tid];
  if (tid < 16) sb1[tid] = (tid < 15) ? bd1[tid] : 0.f;
  if (tid < 16) sb2[tid] = (tid < 8)  ? bd2[tid] : 0.f;

  const int w    = tid >> 5;               // wave id within block (0..7)
  const int lane = tid & 31;
  const int tile = blockIdx.x * 8 + w;     // 0..255
  const int row0 = tile * 16;

  // z tile: cols 0..15 = h chunk, 16..18 = one-hot(code), 19 = pad
  for (int i = lane; i < 320; i += 32) {
    int r = i / 20, k = i % 20;
    float v = (k < 16) ? h[(row0 + r) * 16 + k] : (((k - 16) == code) ? 1.f : 0.f);
    zs[w][r][k] = v;
  }
  __syncthreads();

  const int l15   = lane & 15;             // A row index / B,D column index
  const int off   = (lane < 16) ? 0 : 2;   // K sub-offset per ISA A/B layout
  const int mbase = (lane < 16) ? 0 : 8;   // D rows held by this lane

  // ---- layer 1: z[16x20] @ Wd1[20x16] ----
  v8f acc = {0.f, 0.f, 0.f, 0.f, 0.f, 0.f, 0.f, 0.f};
  #pragma unroll
  for (int kb = 0; kb < 5; ++kb) {
    int k0 = kb * 4 + off;
    v2f a; a.x = zs[w][l15][k0]; a.y = zs[w][l15][k0 + 1];
    v2f b; b.x = sW1[k0][l15];   b.y = sW1[k0 + 1][l15];
    acc = __builtin_amdgcn_wmma_f32_16x16x4_f32(false, a, false, b, (short)0, acc, false, false);
  }
  #pragma unroll
  for (int i = 0; i < 8; ++i)
    hsm[w][mbase + i][l15] = leaky(acc[i] + sb1[l15]);
  __syncthreads();

  // ---- layer 2: h1[16x16] @ Wd2[16x16] (cols>=8 padded zero) ----
  v8f acc2 = {0.f, 0.f, 0.f, 0.f, 0.f, 0.f, 0.f, 0.f};
  #pragma unroll
  for (int kb = 0; kb < 4; ++kb) {
    int k0 = kb * 4 + off;
    v2f a; a.x = hsm[w][l15][k0]; a.y = hsm[w][l15][k0 + 1];
    v2f b; b.x = sW2[k0][l15];    b.y = sW2[k0 + 1][l15];
    acc2 = __builtin_amdgcn_wmma_f32_16x16x4_f32(false, a, false, b, (short)0, acc2, false, false);
  }
  __syncthreads();
  #pragma unroll
  for (int i = 0; i < 8; ++i) {
    float v = leaky(acc2[i] + sb2[l15]);
    hsm[w][mbase + i][l15] = (l15 < 8) ? v : 0.f;
  }
  __syncthreads();

  // ---- layer 3: h2[16x8] @ Wd3[8x1] ----
  v8f acc3 = {0.f, 0.f, 0.f, 0.f, 0.f, 0.f, 0.f, 0.f};
  #pragma unroll
  for (int kb = 0; kb < 2; ++kb) {
    int k0 = kb * 4 + off;
    v2f a; a.x = hsm[w][l15][k0]; a.y = hsm[w][l15][k0 + 1];
    v2f b; b.x = (l15 == 0) ? sW3[k0] : 0.f; b.y = (l15 == 0) ? sW3[k0 + 1] : 0.f;
    acc3 = __builtin_amdgcn_wmma_f32_16x16x4_f32(false, a, false, b, (short)0, acc3, false, false);
  }
  float b3 = bd3[0];
  if (l15 == 0) {             // lanes 0 and 16 hold column 0 (rows 0..7 / 8..15)
    #pragma unroll
    for (int i = 0; i < 8; ++i)
      aout[row0 + mbase + i] = tanhf(acc3[i] + b3);
  }
}

// -------- K2: bandwidth-dominant GEMV, K split 8 ways, NT b128 loads --------
// a0 chunk staged into LDS with GLOBAL_LOAD_ASYNC_TO_LDS (ASYNCcnt path).
__global__ void k_gemv_partial(const float* __restrict__ a0, const float* __restrict__ W1,
                               float* __restrict__ part) {
  __shared__ float sa[KCHUNK];
  const int ks = blockIdx.y;
  const int kbase = ks * KCHUNK;

  // async global->LDS stage of this K-chunk of a0 (2 x B32 per thread)
  #pragma unroll
  for (int j = 0; j < KCHUNK / 256; ++j) {
    int idx = threadIdx.x + j * 256;
    uint32_t lds_off  = (uint32_t)(uintptr_t)(__attribute__((address_space(3))) float*)&sa[idx];
    uint32_t byte_off = (uint32_t)(kbase + idx) * 4u;
    asm volatile("global_load_async_to_lds_b32 %0, %1, %2"
                 :: "v"(lds_off), "v"(byte_off), "s"(a0) : "memory");
  }
  asm volatile("s_wait_asynccnt 0" ::: "memory");
  __syncthreads();

  const int c4 = blockIdx.x * 256 + threadIdx.x;   // float4 column group
  const v4f* W = (const v4f*)W1;
  v4f acc = {0.f, 0.f, 0.f, 0.f};
  #pragma unroll 16
  for (int k = 0; k < KCHUNK; ++k) {
    float a = sa[k];
    v4f wv = __builtin_nontemporal_load(&W[(size_t)(kbase + k) * COLS4 + c4]);
    acc += a * wv;
  }
  ((v4f*)part)[(size_t)ks * COLS4 + c4] = acc;
}

__global__ void k_gemv_reduce(const float* __restrict__ part, const float* __restrict__ b1,
                              const float* __restrict__ x, const float* __restrict__ Ws02,
                              const float* __restrict__ bs02, float* __restrict__ h) {
  const int c4 = blockIdx.x * 256 + threadIdx.x;
  v4f acc = ((const v4f*)part)[c4];
  #pragma unroll
  for (int s = 1; s < KSPLIT; ++s) acc += ((const v4f*)part)[(size_t)s * COLS4 + c4];
  float xv = x[0];
  acc += ((const v4f*)b1)[c4];
  acc += xv * ((const v4f*)Ws02)[c4];
  acc += ((const v4f*)bs02)[c4];
  ((v4f*)h)[c4] = acc;
}

// -------- K4: h2 = a1@W2 + a0@Ws13 + biases, then scalar DAN(code 2) --------
__global__ void k_layer2_final(const float* __restrict__ a0, const float* __restrict__ a1,
                               const float* __restrict__ W2, const float* __restrict__ b2,
                               const float* __restrict__ Ws13, const float* __restrict__ bs13,
                               const float* __restrict__ Wd1, const float* __restrict__ bd1,
                               const float* __restrict__ Wd2, const float* __restrict__ bd2,
                               const float* __restrict__ Wd3, const float* __restrict__ bd3,
                               float* __restrict__ out) {
  __shared__ float red[16][17];
  __shared__ float s_h2[16];
  const int t = threadIdx.x;
  const int c = t & 15, g = t >> 4;
  float p = 0.f;
  const int k0 = g * 256;
  for (int k = k0; k < k0 + 256; ++k)
    p += a1[k] * W2[k * 16 + c] + a0[k] * Ws13[k * 16 + c];
  red[c][g] = p;
  __syncthreads();
  if (t < 16) {
    float s = b2[t] + bs13[t];
    for (int gg = 0; gg < 16; ++gg) s += red[t][gg];
    s_h2[t] = s;
  }
  __syncthreads();
  if (t == 0) {
    float z[19];
    for (int i = 0; i < 16; ++i) z[i] = s_h2[i];
    z[16] = 0.f; z[17] = 0.f; z[18] = 1.f;           // one-hot code 2
    float h1v[15];
    for (int n = 0; n < 15; ++n) {
      float s = bd1[n];
      for (int k = 0; k < 19; ++k) s += z[k] * Wd1[k * 15 + n];
      h1v[n] = leaky(s);
    }
    float h2v[8];
    for (int n = 0; n < 8; ++n) {
      float s = bd2[n];
      for (int k = 0; k < 15; ++k) s += h1v[k] * Wd2[k * 8 + n];
      h2v[n] = leaky(s);
    }
    float o = bd3[0];
    for (int k = 0; k < 8; ++k) o += h2v[k] * Wd3[k];
    out[0] = tanhf(o);
  }
}

extern "C" void kernel_launch(void* const* d_in, const int* in_sizes, int n_in,
                              void* d_out, int out_size, void* d_ws, size_t ws_size,
                              hipStream_t stream) {
  (void)in_sizes; (void)n_in; (void)out_size; (void)ws_size;
  const float* x    = (const float*)d_in[0];
  const float* W0   = (const float*)d_in[1];
  const float* b0   = (const float*)d_in[2];
  const float* W1   = (const float*)d_in[3];
  const float* b1   = (const float*)d_in[4];
  const float* W2   = (const float*)d_in[5];
  const float* b2   = (const float*)d_in[6];
  const float* Ws02 = (const float*)d_in[7];
  const float* bs02 = (const float*)d_in[8];
  const float* Ws13 = (const float*)d_in[9];
  const float* bs13 = (const float*)d_in[10];
  const float* Wd1  = (const float*)d_in[11];
  const float* bd1  = (const float*)d_in[12];
  const float* Wd2  = (const float*)d_in[13];
  const float* bd2  = (const float*)d_in[14];
  const float* Wd3  = (const float*)d_in[15];
  const float* bd3  = (const float*)d_in[16];
  float* out = (float*)d_out;

  float* ws   = (float*)d_ws;
  float* hbuf = ws;               // 65536
  float* a0   = hbuf + NCOL;      // 4096
  float* a1   = a0 + 4096;        // 4096
  float* part = a1 + 4096;        // KSPLIT * 65536

  // layer 0: h0 = x*W0 + b0
  k_affine1<<<64, 256, 0, stream>>>(x, W0, b0, hbuf);
  // a0 = DAN(h0, code 0)
  k_dan_batch<<<32, 256, 0, stream>>>(hbuf, a0, Wd1, bd1, Wd2, bd2, Wd3, bd3, 0);
  // h1 = a0 @ W1 (+ b1 + x*Ws02 + bs02): K-split partials + deterministic reduce
  k_gemv_partial<<<dim3(64, KSPLIT), 256, 0, stream>>>(a0, W1, part);
  k_gemv_reduce<<<64, 256, 0, stream>>>(part, b1, x, Ws02, bs02, hbuf);
  // a1 = DAN(h1, code 1)
  k_dan_batch<<<32, 256, 0, stream>>>(hbuf, a1, Wd1, bd1, Wd2, bd2, Wd3, bd3, 1);
  // h2 + final scalar DAN (code 2) -> out
  k_layer2_final<<<1, 256, 0, stream>>>(a0, a1, W2, b2, Ws13, bs13,
                                        Wd1, bd1, Wd2, bd2, Wd3, bd3, out);
}